// DCRNN_87926570483768
// MI455X (gfx1250) — compile-verified
//
#include <hip/hip_runtime.h>
#include <hip/hip_bf16.h>
#include <math.h>

// Problem constants (from reference)
#define NN_    2048
#define T_     128
#define FIN_   64
#define H_     128
#define G_     384      // 3*H
#define E_     32768

typedef __attribute__((ext_vector_type(2))) float v2f;
typedef __attribute__((ext_vector_type(8))) float v8f;

// ---------------------------------------------------------------------------
// Slice x[:, T-1, :] -> xs (N x F_IN); init deg with self-loop count 1.
// ---------------------------------------------------------------------------
__global__ void slice_kernel(const float* __restrict__ x, float* __restrict__ xs,
                             float* __restrict__ deg) {
    int i = blockIdx.x * blockDim.x + threadIdx.x;
    if (i < NN_ * FIN_) {
        int n = i / FIN_, f = i % FIN_;
        xs[i] = x[(n * T_ + (T_ - 1)) * FIN_ + f];
    }
    if (i < NN_) deg[i] = 1.0f;   // self loop
}

__global__ void deg_edge_kernel(const int* __restrict__ dst, float* __restrict__ deg) {
    int e = blockIdx.x * blockDim.x + threadIdx.x;
    if (e < E_) atomicAdd(&deg[dst[e]], 1.0f);
}

__global__ void dinv_kernel(const float* __restrict__ deg, float* __restrict__ dinv) {
    int n = blockIdx.x * blockDim.x + threadIdx.x;
    if (n < NN_) {
        float d = deg[n];
        dinv[n] = (d > 0.0f) ? rsqrtf(d) : 0.0f;
    }
}

// ---------------------------------------------------------------------------
// f32 WMMA GEMM: C[M x N] = A[M x K] @ B (+ bias per column).
// One wave32 computes one 16x16 tile via V_WMMA_F32_16X16X4_F32.
// BT==0: B row-major [K x N].  BT==1: B stored as [N x K] (i.e. C = A @ B^T).
// A fragment (16x4 f32): lanes 0-15 hold (K=0,1) of row M=lane; lanes 16-31
// hold (K=2,3).  B fragment mirrors this per the ISA VGPR layout tables.
// C/D: VGPR i holds row (i + 8*half), col = lane&15.
// ---------------------------------------------------------------------------
template <int BT>
__global__ void __launch_bounds__(128)
wmma_gemm_f32(const float* __restrict__ A, const float* __restrict__ B,
              const float* __restrict__ bias, float* __restrict__ C,
              int M, int N, int K) {
    int wavesPerBlk = blockDim.x >> 5;
    int waveId = blockIdx.x * wavesPerBlk + (threadIdx.x >> 5);
    int tilesN = N >> 4;
    int tileM = waveId / tilesN;
    int tileN = waveId % tilesN;
    if (tileM * 16 >= M) return;            // wave-uniform guard (EXEC all-1 inside)

    int lane = threadIdx.x & 31;
    int half = lane >> 4;                   // 0: K pair (0,1); 1: K pair (2,3)
    int l    = lane & 15;
    int row0 = tileM * 16, col0 = tileN * 16;

    v8f c;
    float binit = bias ? bias[col0 + l] : 0.0f;
#pragma unroll
    for (int i = 0; i < 8; ++i) c[i] = binit;

    for (int kk = 0; kk < K; kk += 4) {
        int ka = kk + 2 * half;
        v2f a, b;
        a.x = A[(row0 + l) * K + ka];
        a.y = A[(row0 + l) * K + ka + 1];
        if (BT) {
            b.x = B[(col0 + l) * K + ka];
            b.y = B[(col0 + l) * K + ka + 1];
        } else {
            b.x = B[(ka)     * N + col0 + l];
            b.y = B[(ka + 1) * N + col0 + l];
        }
        c = __builtin_amdgcn_wmma_f32_16x16x4_f32(false, a, false, b,
                                                  (short)0, c, false, false);
    }
#pragma unroll
    for (int i = 0; i < 8; ++i)
        C[(row0 + i + 8 * half) * N + col0 + l] = c[i];
}

// ---------------------------------------------------------------------------
// GCN aggregation.  out[n][h] = b[h] + y[n][h]*dinv[n]^2  (self loop + bias),
// then edges add y[src][h] * dinv[src]*dinv[dst] into out[dst][h].
// ---------------------------------------------------------------------------
__global__ void agg_self_bias_kernel(const float* __restrict__ y,
                                     const float* __restrict__ dinv,
                                     const float* __restrict__ b,
                                     float* __restrict__ out) {
    int i = blockIdx.x * blockDim.x + threadIdx.x;
    if (i >= NN_ * H_) return;
    int n = i / H_, h = i % H_;
    float d = dinv[n];
    out[i] = b[h] + y[i] * d * d;
}

__global__ void agg_edge_kernel(const float* __restrict__ y,
                                const int* __restrict__ src,
                                const int* __restrict__ dst,
                                const float* __restrict__ dinv,
                                float* __restrict__ out) {
    const int QUADS = H_ / 4;                      // 32 float4 chunks per row
    int i = blockIdx.x * blockDim.x + threadIdx.x; // one thread per (edge, quad)
    if (i >= E_ * QUADS) return;
    int e = i / QUADS;
    int q = (i % QUADS) * 4;
    int s = src[e], d = dst[e];
    float nrm = dinv[s] * dinv[d];
    const float4 v = *(const float4*)&y[s * H_ + q];
    float* o = &out[d * H_ + q];
    atomicAdd(o + 0, v.x * nrm);
    atomicAdd(o + 1, v.y * nrm);
    atomicAdd(o + 2, v.z * nrm);
    atomicAdd(o + 3, v.w * nrm);
}

// ---------------------------------------------------------------------------
// Sequential GRU scan over the node axis (batch collapsed to the single
// surviving t = T-1 row).  gi = y2 @ w_ih^T + b_ih is precomputed (WMMA).
// Per step: gh = h @ w_hh^T (+b_hh), gates, h update, out[n] = h . fc_w + fc_b.
// One workgroup, 384 threads (one per gate output), w_hh staged in LDS with a
// padded row stride for conflict-light ds_load_b128.
// ---------------------------------------------------------------------------
#define WROW 132   // 128 + 4 pad: 16B-aligned rows, 4-bank lane stride

__global__ void __launch_bounds__(G_)
gru_scan_kernel(const float* __restrict__ gi, const float* __restrict__ w_hh,
                const float* __restrict__ b_hh, const float* __restrict__ fc_w,
                const float* __restrict__ fc_b, float* __restrict__ out) {
    extern __shared__ float smem[];
    float* whh   = smem;                 // G_ * WROW
    float* hbuf  = whh + G_ * WROW;      // H_
    float* ghbuf = hbuf + H_;            // G_
    float* pbuf  = ghbuf + G_;           // H_

    const int tid = threadIdx.x;

    // Stage w_hh (row-major [G_ x H_]) into padded LDS rows.
    for (int idx = tid; idx < G_ * H_; idx += G_) {
        int g = idx >> 7, k = idx & (H_ - 1);
        whh[g * WROW + k] = w_hh[idx];
    }
    if (tid < H_) hbuf[tid] = 0.0f;
    float bh = b_hh[tid];
    float fw = (tid < H_) ? fc_w[tid] : 0.0f;
    float fb = fc_b[0];
    __syncthreads();

    for (int n = 0; n < NN_; ++n) {
        // gh[tid] = b_hh[tid] + sum_k h[k] * w_hh[tid][k]
        float acc = bh;
        const float* wr = &whh[tid * WROW];
#pragma unroll 8
        for (int k = 0; k < H_; k += 4) {
            float4 hv = *(const float4*)&hbuf[k];   // broadcast read
            float4 wv = *(const float4*)&wr[k];     // ds_load_b128
            acc = fmaf(hv.x, wv.x, acc);
            acc = fmaf(hv.y, wv.y, acc);
            acc = fmaf(hv.z, wv.z, acc);
            acc = fmaf(hv.w, wv.w, acc);
        }
        ghbuf[tid] = acc;
        __syncthreads();

        if (tid < H_) {
            const float* gin_ = &gi[(size_t)n * G_];
            float r  = 1.0f / (1.0f + expf(-(gin_[tid]            + ghbuf[tid])));
            float z  = 1.0f / (1.0f + expf(-(gin_[H_ + tid]       + ghbuf[H_ + tid])));
            float nv = tanhf(gin_[2 * H_ + tid] + r * ghbuf[2 * H_ + tid]);
            float hnew = (1.0f - z) * nv + z * hbuf[tid];
            hbuf[tid] = hnew;
            pbuf[tid] = hnew * fw;
        }
        __syncthreads();

        if (tid < 32) {
            float s = pbuf[tid] + pbuf[tid + 32] + pbuf[tid + 64] + pbuf[tid + 96];
            for (int off = 16; off; off >>= 1) s += __shfl_down(s, off, 32);
            if (tid == 0) out[n] = s + fb;
        }
        // pbuf is only rewritten after the next iteration's first barrier,
        // which the reducing wave also reaches — no extra barrier needed.
    }
}

// ---------------------------------------------------------------------------
extern "C" void kernel_launch(void* const* d_in, const int* in_sizes, int n_in,
                              void* d_out, int out_size, void* d_ws, size_t ws_size,
                              hipStream_t stream) {
    const float* x     = (const float*)d_in[0];
    const int*   eidx  = (const int*)d_in[1];      // (2, E) row-major int32
    const float* W1    = (const float*)d_in[2];
    const float* b1    = (const float*)d_in[3];
    const float* W2    = (const float*)d_in[4];
    const float* b2    = (const float*)d_in[5];
    const float* w_ih  = (const float*)d_in[6];    // (384, 128)
    const float* w_hh  = (const float*)d_in[7];    // (384, 128)
    const float* b_ih  = (const float*)d_in[8];
    const float* b_hh  = (const float*)d_in[9];
    const float* fc_w  = (const float*)d_in[10];   // (1, 128)
    const float* fc_b  = (const float*)d_in[11];
    float* out = (float*)d_out;

    const int* src = eidx;          // edge_index[0]
    const int* dst = eidx + E_;     // edge_index[1]

    // Workspace carve-up (floats)
    float* ws   = (float*)d_ws;
    float* xs   = ws;                 ws += NN_ * FIN_;   // 2048x64
    float* deg  = ws;                 ws += NN_;
    float* dinv = ws;                 ws += NN_;
    float* t1   = ws;                 ws += NN_ * H_;     // post-GEMM1
    float* a1   = ws;                 ws += NN_ * H_;     // post-agg1
    float* t2   = ws;                 ws += NN_ * H_;
    float* y2   = ws;                 ws += NN_ * H_;
    float* gi   = ws;                 ws += NN_ * G_;     // 2048x384
    (void)ws_size; (void)in_sizes; (void)n_in; (void)out_size;

    // 1) slice + degree
    slice_kernel<<<(NN_ * FIN_ + 255) / 256, 256, 0, stream>>>(x, xs, deg);
    deg_edge_kernel<<<(E_ + 255) / 256, 256, 0, stream>>>(dst, deg);
    dinv_kernel<<<(NN_ + 255) / 256, 256, 0, stream>>>(deg, dinv);

    // 2) GCN conv 1: t1 = xs @ W1 ; a1 = aggregate(t1) + b1
    {
        int waves = (NN_ / 16) * (H_ / 16);       // 1024
        wmma_gemm_f32<0><<<waves / 4, 128, 0, stream>>>(xs, W1, nullptr, t1,
                                                        NN_, H_, FIN_);
    }
    agg_self_bias_kernel<<<(NN_ * H_ + 255) / 256, 256, 0, stream>>>(t1, dinv, b1, a1);
    agg_edge_kernel<<<(E_ * (H_ / 4) + 255) / 256, 256, 0, stream>>>(t1, src, dst, dinv, a1);

    // 3) GCN conv 2: t2 = a1 @ W2 ; y2 = aggregate(t2) + b2
    {
        int waves = (NN_ / 16) * (H_ / 16);
        wmma_gemm_f32<0><<<waves / 4, 128, 0, stream>>>(a1, W2, nullptr, t2,
                                                        NN_, H_, H_);
    }
    agg_self_bias_kernel<<<(NN_ * H_ + 255) / 256, 256, 0, stream>>>(t2, dinv, b2, y2);
    agg_edge_kernel<<<(E_ * (H_ / 4) + 255) / 256, 256, 0, stream>>>(t2, src, dst, dinv, y2);

    // 4) GRU input gates for all steps at once: gi = y2 @ w_ih^T + b_ih
    {
        int waves = (NN_ / 16) * (G_ / 16);       // 3072
        wmma_gemm_f32<1><<<waves / 4, 128, 0, stream>>>(y2, w_ih, b_ih, gi,
                                                        NN_, G_, H_);
    }

    // 5) Sequential GRU scan + fused final FC
    size_t shmem = (size_t)(G_ * WROW + H_ + G_ + H_) * sizeof(float); // ~205 KB
    gru_scan_kernel<<<1, G_, shmem, stream>>>(gi, w_hh, b_hh, fc_w, fc_b, out);
}